// WeightedSum_42271068127242
// MI455X (gfx1250) — compile-verified
//
#include <hip/hip_runtime.h>

typedef float v2f __attribute__((ext_vector_type(2)));
typedef float v8f __attribute__((ext_vector_type(8)));

#define B_DIM 64
#define C_DIM 512
#define HW    3136            // 56*56
#define PIX_PER_BLOCK 64      // 4 waves * 16-pixel WMMA tile
#define BLOCKS_PER_IMAGE 49   // 3136 / 64

// ---------------------------------------------------------------------------
// Kernel 1: image[b,p] = dot(batch[b,p,:], a[b,:]) via V_WMMA_F32_16X16X4_F32
// One wave32 computes a 16-pixel tile; K loop steps 4 channels per WMMA.
// B-matrix columns are all replicated with a[k..k+3], so every column of the
// 16x16 accumulator equals the desired dot products.
// ---------------------------------------------------------------------------
__global__ __launch_bounds__(128)
void weighted_sum_wmma(const float* __restrict__ batch,
                       const float* __restrict__ a_batch,
                       float* __restrict__ out)
{
    __shared__ float s_a[C_DIM];

    const int tid  = threadIdx.x;
    const int b    = blockIdx.x / BLOCKS_PER_IMAGE;
    const int blk  = blockIdx.x % BLOCKS_PER_IMAGE;

    // Stage a_batch[b,:] (2 KB) into LDS: 128 threads x float4 = 512 floats.
    {
        const float4* src = (const float4*)(a_batch + (size_t)b * C_DIM);
        ((float4*)s_a)[tid] = src[tid];
    }
    __syncthreads();

    const int wave = tid >> 5;
    const int lane = tid & 31;
    const int half = lane >> 4;   // 0 -> K = k,k+1 ; 1 -> K = k+2,k+3
    const int row  = lane & 15;   // M (pixel within tile)

    const int p0 = blk * PIX_PER_BLOCK + wave * 16;
    const float* Arow =
        batch + ((size_t)b * HW + (size_t)(p0 + row)) * C_DIM + half * 2;
    const float* Bvec = s_a + half * 2;

    v8f acc = {};
    #pragma unroll 4
    for (int k = 0; k < C_DIM; k += 4) {
        // A: lane L -> X[row][k + 2*half .. +1]
        v2f a = *(const v2f*)(Arow + k);
        // B: replicate a[k + 2*half .. +1] across all 16 columns
        v2f bv = *(const v2f*)(Bvec + k);
        // 8 args: (neg_a, A, neg_b, B, c_mod, C, reuse_a, reuse_b)
        acc = __builtin_amdgcn_wmma_f32_16x16x4_f32(
            false, a, false, bv, (short)0, acc, false, false);
    }

    // Every column of acc is identical:
    //   lane 0  holds pixels p0+0..7  in acc[0..7]
    //   lane 16 holds pixels p0+8..15 in acc[0..7]
    if (row == 0) {
        float* dst = out + (size_t)b * HW + p0 + half * 8;
        float4 lo = {acc[0], acc[1], acc[2], acc[3]};
        float4 hi = {acc[4], acc[5], acc[6], acc[7]};
        ((float4*)dst)[0] = lo;
        ((float4*)dst)[1] = hi;
    }
}

// ---------------------------------------------------------------------------
// Kernel 2: per-image max over 3136 values, then in-place scale by 1/max.
// One 256-thread (8-wave) block per image; wave32 shuffle + LDS reduction.
// ---------------------------------------------------------------------------
__global__ __launch_bounds__(256)
void normalize_max(float* __restrict__ img)
{
    __shared__ float s_red[8];
    float* p = img + (size_t)blockIdx.x * HW;
    const int tid = threadIdx.x;

    float m = -1e30f;
    for (int i = tid; i < HW; i += 256) m = fmaxf(m, p[i]);

    // intra-wave reduction (wave32)
    #pragma unroll
    for (int off = 16; off > 0; off >>= 1)
        m = fmaxf(m, __shfl_xor(m, off, 32));
    if ((tid & 31) == 0) s_red[tid >> 5] = m;
    __syncthreads();

    if (tid < 8) {
        m = s_red[tid];
        #pragma unroll
        for (int off = 4; off > 0; off >>= 1)
            m = fmaxf(m, __shfl_xor(m, off, 32)); // lanes 0..7 stay in-group
        if (tid == 0) s_red[0] = m;
    }
    __syncthreads();

    const float inv = 1.0f / s_red[0];
    for (int i = tid; i < HW; i += 256) p[i] *= inv;
}

extern "C" void kernel_launch(void* const* d_in, const int* in_sizes, int n_in,
                              void* d_out, int out_size, void* d_ws, size_t ws_size,
                              hipStream_t stream)
{
    const float* batch   = (const float*)d_in[0];
    const float* a_batch = (const float*)d_in[1];
    float* out = (float*)d_out;

    weighted_sum_wmma<<<dim3(B_DIM * BLOCKS_PER_IMAGE), dim3(128), 0, stream>>>(
        batch, a_batch, out);
    normalize_max<<<dim3(B_DIM), dim3(256), 0, stream>>>(out);
}